// FrozenKernel_75419625718629
// MI455X (gfx1250) — compile-verified
//
#include <hip/hip_runtime.h>

typedef __attribute__((ext_vector_type(2))) float v2f;
typedef __attribute__((ext_vector_type(4))) float v4f;
typedef __attribute__((ext_vector_type(8))) float v8f;

#define EDGES  16
#define HID    128
#define HS     132      // padded LDS stride for h (bank-conflict-free A frags)
#define PDIM   1024     // columns per path (32 w * 32 u)
#define W2COLS 5120
#define DDIM   128

// constants
#define K_N0   0.125f          // 1/sqrt(2*32)
#define K_N1   0.10206207262f  // 1/sqrt(3*32)
#define K_C01  0.07216878365f  // n0/sqrt(3)
#define K_C11E 0.07216878365f  // n1/sqrt(2) (== n0/sqrt(3))
#define K_S    0.17677669529f  // 1/sqrt(32)
#define SQRT3F 1.73205080757f

// One path of R = h(16x128) @ W2[:, p*1024 : (p+1)*1024] + b2, via f32 WMMA.
__device__ __forceinline__ void gemm_path(const float* __restrict__ hb,
                                          const float* __restrict__ W2,
                                          const float* __restrict__ b2,
                                          float* __restrict__ buf,
                                          int path, int wave, int lane)
{
    const int m    = lane & 15;
    const int half = lane >> 4;
    const float* W2p = W2 + path * PDIM;
    for (int t = 0; t < 8; ++t) {
        const int jt  = wave + t * 8;          // 0..63 column tiles
        const int col = jt * 16 + m;
        v8f acc = {0.f, 0.f, 0.f, 0.f, 0.f, 0.f, 0.f, 0.f};
#pragma unroll 4
        for (int kk = 0; kk < 32; ++kk) {
            const int k = kk * 4 + half * 2;   // A/B: K = kbase + 2*half (+1 for .y)
            v2f a = *(const v2f*)(hb + m * HS + k);              // ds_load_b64
            v2f b;
            b.x = W2p[(size_t)k       * W2COLS + col];
            b.y = W2p[(size_t)(k + 1) * W2COLS + col];
            acc = __builtin_amdgcn_wmma_f32_16x16x4_f32(
                false, a, false, b, (short)0, acc, false, false);
        }
        const float bb = b2[path * PDIM + jt * 16 + m];
#pragma unroll
        for (int q = 0; q < 8; ++q)            // D row = q + 8*half
            buf[(q + half * 8) * PDIM + jt * 16 + m] = acc[q] + bb;
    }
}

__global__ void __launch_bounds__(256, 1)
frozen_kernel(const float* __restrict__ r,  const float* __restrict__ W1,
              const float* __restrict__ b1, const float* __restrict__ W2,
              const float* __restrict__ b2, const float* __restrict__ lw0,
              const float* __restrict__ lw1, float* __restrict__ out, int N)
{
    extern __shared__ float sm[];
    float* hb   = sm;                          // [16][132]
    float* bufA = hb   + EDGES * HS;           // [16][1024]
    float* bufB = bufA + EDGES * PDIM;         // [16][1024]
    float* bufC = bufB + EDGES * PDIM;         // [16][1024]
    float* slw0 = bufC + EDGES * PDIM;         // 1024 (pre-scaled by 1/sqrt(32))
    float* slw1 = slw0 + 1024;                 // 1024
    float* sy   = slw1 + 1024;                 // [16][3]  Y1 = sqrt3*unit
    float* sMt  = sy   + EDGES * 3;            // [16][9]  M[j][i] = sum_k Y[k] eps[i,k,j]
    float* smk  = sMt  + EDGES * 9;            // [16] mask
    float* sr   = smk  + EDGES;                // [16] radius

    const int tid  = threadIdx.x;
    const int lane = tid & 31;
    const int wave = tid >> 5;
    const int e0   = blockIdx.x * EDGES;

    // -------- phase 0: per-edge geometry + weights into LDS --------
    if (tid < EDGES) {
        const int n = e0 + tid;
        float x = 0.f, y = 0.f, z = 0.f;
        if (n < N) { x = r[3*n]; y = r[3*n+1]; z = r[3*n+2]; }
        const float rad = sqrtf(x*x + y*y + z*z);
        const bool  msk = rad > 0.1f;
        const float inv = msk ? (SQRT3F / rad) : 0.f;
        const float Y0 = x * inv, Y1v = y * inv, Y2 = z * inv;
        sr[tid]  = rad;
        smk[tid] = msk ? 1.f : 0.f;
        sy[tid*3+0] = Y0; sy[tid*3+1] = Y1v; sy[tid*3+2] = Y2;
        float* Mt = sMt + tid * 9;             // rows j, cols i
        Mt[0] = 0.f;  Mt[1] = Y2;   Mt[2] = -Y1v;
        Mt[3] = -Y2;  Mt[4] = 0.f;  Mt[5] = Y0;
        Mt[6] = Y1v;  Mt[7] = -Y0;  Mt[8] = 0.f;
    }
    for (int i = tid; i < 1024; i += 256) {
        slw0[i] = K_S * lw0[i];
        slw1[i] = K_S * lw1[i];
    }
    __syncthreads();
    for (int idx = tid; idx < EDGES * HID; idx += 256) {
        const int e = idx >> 7, k = idx & 127;
        const float hv = sr[e] * W1[k] + b1[k];
        hb[e * HS + k] = hv > 0.f ? hv : 0.f;
    }
    __syncthreads();

    // -------- pass 1: R0a (p0) -> bufA, R1b (p3) -> bufB --------
    gemm_path(hb, W2, b2, bufA, 0, wave, lane);
    gemm_path(hb, W2, b2, bufB, 3, wave, lane);
    __syncthreads();

    // rows 0..31 of each edge: [ K00 | K01 ]
    {
        const int cbase = lane * 4;
        for (int it = 0; it < 64; ++it) {
            const int rId = wave + it * 8;     // 0..511
            const int e = rId >> 5, w = rId & 31;
            const int n = e0 + e;
            if (n >= N) continue;              // uniform per wave
            const float msk = smk[e];
            v4f v;
            if (cbase < 32) {                  // lanes 0..7 entirely in K00
#pragma unroll
                for (int c = 0; c < 4; ++c) {
                    const int u = cbase + c;
                    const float kv = K_N0 * bufA[e * PDIM + w * 32 + u];
                    v[c] = (msk != 0.f) ? kv : slw0[w * 32 + u];
                }
            } else {                           // K01: col = 32 + u*3 + i
#pragma unroll
                for (int c = 0; c < 4; ++c) {
                    const int t = cbase - 32 + c;
                    const int u = t / 3, i = t - u * 3;
                    v[c] = msk * K_C01 * bufB[e * PDIM + w * 32 + u] * sy[e * 3 + i];
                }
            }
            *(v4f*)(out + ((size_t)n * DDIM + w) * DDIM + cbase) = v;
        }
    }
    __syncthreads();

    // -------- pass 2: R1a (p2) -> bufA, R0b (p1) -> bufB, R1c (p4) -> bufC --------
    gemm_path(hb, W2, b2, bufA, 2, wave, lane);
    gemm_path(hb, W2, b2, bufB, 1, wave, lane);
    gemm_path(hb, W2, b2, bufC, 4, wave, lane);
    __syncthreads();

    // rows 32..127 of each edge: [ K10 | K11 ]
    {
        const int cbase = lane * 4;
        for (int it = 0; it < 192; ++it) {
            const int rId = wave + it * 8;     // 0..1535
            const int e = rId / 96;
            const int q = rId - e * 96;        // row - 32
            const int w = q / 3, j = q - w * 3;
            const int n = e0 + e;
            if (n >= N) continue;
            const float msk = smk[e];
            v4f v;
            if (cbase < 32) {                  // K10: row = 32 + w*3 + j, col = u
                const float yj = sy[e * 3 + j];
#pragma unroll
                for (int c = 0; c < 4; ++c) {
                    const int u = cbase + c;
                    v[c] = msk * K_N1 * bufA[e * PDIM + w * 32 + u] * yj;
                }
            } else {                           // K11: col = 32 + u*3 + i
#pragma unroll
                for (int c = 0; c < 4; ++c) {
                    const int t = cbase - 32 + c;
                    const int u = t / 3, i = t - u * 3;
                    const float diag = (i == j) ? 1.f : 0.f;
                    const float k1 = K_N1  * bufB[e * PDIM + w * 32 + u] * diag
                                   + K_C11E * bufC[e * PDIM + w * 32 + u] * sMt[e * 9 + j * 3 + i];
                    v[c] = (msk != 0.f) ? k1 : slw1[w * 32 + u] * diag;
                }
            }
            *(v4f*)(out + ((size_t)n * DDIM + 32 + q) * DDIM + cbase) = v;
        }
    }
}

extern "C" void kernel_launch(void* const* d_in, const int* in_sizes, int n_in,
                              void* d_out, int out_size, void* d_ws, size_t ws_size,
                              hipStream_t stream) {
    const float* r   = (const float*)d_in[0];
    const float* W1  = (const float*)d_in[1];
    const float* b1  = (const float*)d_in[2];
    const float* W2  = (const float*)d_in[3];
    const float* b2  = (const float*)d_in[4];
    const float* lw0 = (const float*)d_in[5];
    const float* lw1 = (const float*)d_in[6];
    float* out = (float*)d_out;

    const int N    = in_sizes[0] / 3;
    const int grid = (N + EDGES - 1) / EDGES;
    const size_t shmem =
        (size_t)(EDGES * HS + 3 * EDGES * PDIM + 2 * 1024 +
                 EDGES * 3 + EDGES * 9 + EDGES + EDGES) * sizeof(float); // ~214 KB

    hipFuncSetAttribute(reinterpret_cast<const void*>(frozen_kernel),
                        hipFuncAttributeMaxDynamicSharedMemorySize, (int)shmem);
    frozen_kernel<<<grid, 256, shmem, stream>>>(r, W1, b1, W2, b2, lw0, lw1, out, N);
}